// NMS_coords_62560493634044
// MI455X (gfx1250) — compile-verified
//
#include <hip/hip_runtime.h>
#include <stdint.h>

#define TOPK  8
#define MDIM  32
#define HW    19200      // 120*160, divisible by 256
#define BLK   256
#define DEPTH 4          // LDS stage buffers (3 stages in flight)

__global__ __launch_bounds__(BLK) void nms_proj_kernel(
    const float* __restrict__ coords,   // [N, M, 3, H, W]
    const float* __restrict__ Pmat,     // [N, 3, 4]
    int* __restrict__ out)              // [N, H, W, TOPK]
{
    __shared__ float buf[DEPTH][3][BLK];

    const int tid = threadIdx.x;
    const int q0  = blockIdx.x * BLK;       // first flat pixel of this block
    const int n   = q0 / HW;                // block never straddles n
    const int pix = q0 - n * HW;

    // global base for this thread's pixel, element ((n*M+0)*3+0)*HW + pix + tid
    const float* gp = coords + (size_t)n * (MDIM * 3) * (size_t)HW + pix + tid;

    // 32-bit LDS byte offset of this thread's (stage0, c0) slot.
    // Generic LDS pointers are {aperture_hi32, lds_offset}; truncation gives the DS offset.
    const uint32_t lds0 = (uint32_t)(uintptr_t)&buf[0][0][tid];

    // uniform camera matrix
    const float* Pr = Pmat + n * 12;
    const float p00=Pr[0], p01=Pr[1], p02=Pr[2],  p03=Pr[3];
    const float p10=Pr[4], p11=Pr[5], p12=Pr[6],  p13=Pr[7];
    const float p20=Pr[8], p21=Pr[9], p22=Pr[10], p23=Pr[11];

    // Issue 3 async global->LDS B32 copies for stage s (this thread's x,y,z).
    auto issue = [&](int s) {
        const float* g = gp + (size_t)(s * 3) * HW;
        const uint32_t l = lds0 + (uint32_t)(s & (DEPTH - 1)) * (3u * BLK * 4u);
#pragma unroll
        for (int c = 0; c < 3; ++c) {
            asm volatile("global_load_async_to_lds_b32 %0, %1, off"
                         :
                         : "v"(l + (uint32_t)c * (BLK * 4u)),
                           "v"(g + (size_t)c * HW)
                         : "memory");
        }
    };

    // 3-deep prefetch pipeline (9 async B32 ops in flight, ASYNCcnt-tracked)
    issue(0); issue(1); issue(2);

    float kx[TOPK], ky[TOPK];
    int   ki[TOPK];
#pragma unroll
    for (int t = 0; t < TOPK; ++t) { kx[t] = 1e9f; ky[t] = 1e9f; ki[t] = 0; }
    int cnt = 0;

    for (int m = 0; m < MDIM; ++m) {
        // stage m complete when outstanding async ops <= 3*(stages still in flight)
        if (m < MDIM - 2)      asm volatile("s_wait_asynccnt 6" ::: "memory");
        else if (m == MDIM - 2) asm volatile("s_wait_asynccnt 3" ::: "memory");
        else                    asm volatile("s_wait_asynccnt 0" ::: "memory");

        const int sb = m & (DEPTH - 1);
        const float x = buf[sb][0][tid];
        const float y = buf[sb][1][tid];
        const float z = buf[sb][2][tid];

        if (m + 3 < MDIM) issue(m + 3);   // writes slot (m+3)&3 != sb

        // projection: X = P[:, :3] @ [x,y,z] + P[:,3]
        const float X0 = fmaf(p02, z, fmaf(p01, y, p00 * x)) + p03;
        const float X1 = fmaf(p12, z, fmaf(p11, y, p10 * x)) + p13;
        const float X2 = fmaf(p22, z, fmaf(p21, y, p20 * x)) + p23;
        const float zc = fmaxf(X2, 1e-6f);
        const float u  = X0 / zc;
        const float v  = X1 / zc;

        // greedy NMS step (branch-free, all state in VGPRs)
        bool sup = false;
#pragma unroll
        for (int t = 0; t < TOPK; ++t) {
            const float dx = kx[t] - u;
            const float dy = ky[t] - v;
            sup = sup || (dx * dx + dy * dy <= 4.0f);
        }
        const bool keep = (!sup) && (cnt < TOPK);
#pragma unroll
        for (int t = 0; t < TOPK; ++t) {
            const bool wr = keep && (cnt == t);
            kx[t] = wr ? u : kx[t];
            ky[t] = wr ? v : ky[t];
            ki[t] = wr ? m : ki[t];
        }
        cnt += keep ? 1 : 0;
    }

    int* op = out + (size_t)(q0 + tid) * TOPK;
    int4 lo = make_int4(ki[0], ki[1], ki[2], ki[3]);
    int4 hi = make_int4(ki[4], ki[5], ki[6], ki[7]);
    *reinterpret_cast<int4*>(op)     = lo;
    *reinterpret_cast<int4*>(op + 4) = hi;
}

extern "C" void kernel_launch(void* const* d_in, const int* in_sizes, int n_in,
                              void* d_out, int out_size, void* d_ws, size_t ws_size,
                              hipStream_t stream) {
    const float* coords = (const float*)d_in[0];   // [N,32,3,120,160] f32
    const float* P      = (const float*)d_in[1];   // [N,3,4] f32
    int* out            = (int*)d_out;             // [N,120,160,8] i32

    const int N       = in_sizes[0] / (MDIM * 3 * HW);
    const int nblocks = (N * HW) / BLK;            // HW % BLK == 0

    nms_proj_kernel<<<nblocks, BLK, 0, stream>>>(coords, P, out);
}